// Mixer_20332375179583
// MI455X (gfx1250) — compile-verified
//
#include <hip/hip_runtime.h>
#include <hip/hip_bf16.h>

typedef __attribute__((ext_vector_type(2))) float v2f;
typedef __attribute__((ext_vector_type(8))) float v8f;

#define B_ 16
#define C_ 256
#define P_ 16384            // 128*128 pixels
#define PBLK 256            // pixels per block in gemm (8 waves x 2 tiles x 16)
#define NPBLK (P_ / PBLK)   // 64 pixel-blocks per (b, o_tile)
#define EPS_ 1e-5f

// ---------------- Kernel 1: per-(b,c) mean/rstd of an activation tensor ----
__global__ __launch_bounds__(256) void stats_x(const float* __restrict__ X,
                                               float* __restrict__ mean,
                                               float* __restrict__ rstd) {
    __shared__ float ss[256], qq[256];
    const int bc = blockIdx.x;
    const float* p = X + (size_t)bc * P_;
    float s = 0.f, q = 0.f;
    for (int i = threadIdx.x * 4; i < P_; i += 256 * 4) {
        float4 v = *(const float4*)(p + i);
        s += v.x + v.y + v.z + v.w;
        q += v.x * v.x + v.y * v.y + v.z * v.z + v.w * v.w;
    }
    ss[threadIdx.x] = s; qq[threadIdx.x] = q;
    __syncthreads();
    for (int off = 128; off > 0; off >>= 1) {
        if (threadIdx.x < (unsigned)off) {
            ss[threadIdx.x] += ss[threadIdx.x + off];
            qq[threadIdx.x] += qq[threadIdx.x + off];
        }
        __syncthreads();
    }
    if (threadIdx.x == 0) {
        float m = ss[0] * (1.0f / P_);
        float var = qq[0] * (1.0f / P_) - m * m;
        mean[bc] = m;
        rstd[bc] = rsqrtf(var + EPS_);
    }
}

// ---------------- Kernel 2: fold instance-norm into weights + bias ---------
// Ws[b][o][c] = W[o][c]*rstd[b][c];  beff[b][o] = bias[o] - sum_c Ws*mean
__global__ __launch_bounds__(256) void scale_w(const float* __restrict__ W,
                                               const float* __restrict__ bias,
                                               const float* __restrict__ mean,
                                               const float* __restrict__ rstd,
                                               float* __restrict__ Ws,
                                               float* __restrict__ beff) {
    __shared__ float red[256];
    const int b = blockIdx.x >> 8;
    const int o = blockIdx.x & 255;
    const int c = threadIdx.x;
    float val = W[o * C_ + c] * rstd[b * C_ + c];
    Ws[((size_t)(b * C_ + o)) * C_ + c] = val;
    red[c] = val * mean[b * C_ + c];
    __syncthreads();
    for (int off = 128; off > 0; off >>= 1) {
        if (c < off) red[c] += red[c + off];
        __syncthreads();
    }
    if (c == 0) beff[b * C_ + o] = bias[o] - red[0];
}

// ---------------- Kernel 3: WMMA fp32 GEMM + bias + relu (+opt stats) ------
// Y[b][o][p] = relu( sum_c Ws[b][o][c] * X[b][c][p] + beff[b][o] )
// 8 wave32s per block; each wave computes TWO 16x16 output tiles (same A
// fragment, two pixel tiles) via 2x64 V_WMMA_F32_16X16X4_F32 (K = 256).
__global__ __launch_bounds__(256) void gemm16(const float* __restrict__ X,
                                              const float* __restrict__ Ws,
                                              const float* __restrict__ beff,
                                              float* __restrict__ Y,
                                              float* __restrict__ psum,
                                              float* __restrict__ psq,
                                              int do_stats) {
    __shared__ float tile[8][2][16][16];   // [wave][ntile][o-row][pixel]
    __shared__ float ps[16][16], pq[16][16];

    const int tid  = threadIdx.x;
    const int wave = tid >> 5;
    const int lane = tid & 31;
    const int n = lane & 15;   // column (pixel) / row m for A
    const int g = lane >> 4;   // lane-half selects K pair

    const int bid    = blockIdx.x;
    const int p_blk  = bid & (NPBLK - 1);
    const int o_tile = (bid >> 6) & 15;
    const int b      = bid >> 10;

    const int o_base = o_tile * 16;
    const int p0     = p_blk * PBLK + wave * 32 + n;   // first pixel tile
                                                       // second tile = p0+16

    // A row: scaled weights for out-channel m = n (lane 0..15 / 16..31 alias)
    const float* Arow  = Ws + ((size_t)(b * C_ + o_base + n)) * C_;
    const float* Bcol0 = X + (size_t)b * C_ * P_ + p0;

    v8f acc0 = {};
    v8f acc1 = {};
#pragma unroll 4
    for (int k0 = 0; k0 < C_; k0 += 4) {
        const int c0 = k0 + 2 * g;
        v2f a = *(const v2f*)(Arow + c0);          // A[m][c0], A[m][c0+1]
        v2f bb0, bb1;
        const float* r0 = Bcol0 + (size_t)c0 * P_;
        const float* r1 = Bcol0 + (size_t)(c0 + 1) * P_;
        bb0.x = r0[0];  bb0.y = r1[0];             // B[c][p0]
        bb1.x = r0[16]; bb1.y = r1[16];            // B[c][p0+16]
        acc0 = __builtin_amdgcn_wmma_f32_16x16x4_f32(
            false, a, false, bb0, (short)0, acc0, false, false);
        acc1 = __builtin_amdgcn_wmma_f32_16x16x4_f32(
            false, a, false, bb1, (short)0, acc1, false, false);
    }

    // epilogue: bias + relu into LDS tiles (D row = v + 8*g, col = n)
#pragma unroll
    for (int v = 0; v < 8; ++v) {
        const int m = v + 8 * g;
        const float bias = beff[b * C_ + o_base + m];
        tile[wave][0][m][n] = fmaxf(acc0[v] + bias, 0.0f);
        tile[wave][1][m][n] = fmaxf(acc1[v] + bias, 0.0f);
    }
    __syncthreads();

    // coalesced global store + deterministic per-row partial sums
    const int r   = tid >> 4;   // out-channel row 0..15
    const int seg = tid & 15;   // 16-pixel segment 0..15 (256 pixels total)
    const int wv  = seg >> 1;   // source wave
    const int hf  = seg & 1;    // source sub-tile
    float s = 0.f, q = 0.f;
    float vals[16];
#pragma unroll
    for (int j = 0; j < 16; ++j) {
        float val = tile[wv][hf][r][j];
        vals[j] = val;
        s += val; q += val * val;
    }
    const size_t orow = ((size_t)(b * C_ + o_base + r)) * P_ + p_blk * PBLK + seg * 16;
#pragma unroll
    for (int j = 0; j < 4; ++j) {
        *(float4*)(Y + orow + 4 * j) =
            make_float4(vals[4 * j], vals[4 * j + 1], vals[4 * j + 2], vals[4 * j + 3]);
    }

    if (do_stats) {
        ps[r][seg] = s; pq[r][seg] = q;
        __syncthreads();
        if (tid < 16) {
            float S = 0.f, Q = 0.f;
#pragma unroll
            for (int j = 0; j < 16; ++j) { S += ps[tid][j]; Q += pq[tid][j]; }
            const int idx = (b * C_ + o_base + tid) * NPBLK + p_blk;
            psum[idx] = S; psq[idx] = Q;
        }
    }
}

// ---------------- Kernel 4: reduce per-block partials -> mean2/rstd2 -------
__global__ __launch_bounds__(64) void stats_part(const float* __restrict__ psum,
                                                 const float* __restrict__ psq,
                                                 float* __restrict__ mean,
                                                 float* __restrict__ rstd) {
    __shared__ float ss[64], qq[64];
    const int bc = blockIdx.x;
    const int t = threadIdx.x;
    ss[t] = psum[bc * NPBLK + t];
    qq[t] = psq[bc * NPBLK + t];
    __syncthreads();
    for (int off = 32; off > 0; off >>= 1) {
        if (t < off) { ss[t] += ss[t + off]; qq[t] += qq[t + off]; }
        __syncthreads();
    }
    if (t == 0) {
        float m = ss[0] * (1.0f / P_);
        float var = qq[0] * (1.0f / P_) - m * m;
        mean[bc] = m;
        rstd[bc] = rsqrtf(var + EPS_);
    }
}

extern "C" void kernel_launch(void* const* d_in, const int* in_sizes, int n_in,
                              void* d_out, int out_size, void* d_ws, size_t ws_size,
                              hipStream_t stream) {
    const float* x  = (const float*)d_in[0];   // [16,256,128,128]
    const float* w1 = (const float*)d_in[1];   // [256,256]
    const float* b1 = (const float*)d_in[2];   // [256]
    const float* w2 = (const float*)d_in[3];   // [256,256]
    const float* b2 = (const float*)d_in[4];   // [256]
    float* out = (float*)d_out;

    // workspace carve-up (floats)
    float* ws = (float*)d_ws;
    float* mean1 = ws; ws += B_ * C_;                 // 4096
    float* rstd1 = ws; ws += B_ * C_;
    float* mean2 = ws; ws += B_ * C_;
    float* rstd2 = ws; ws += B_ * C_;
    float* beff1 = ws; ws += B_ * C_;
    float* beff2 = ws; ws += B_ * C_;
    float* psum  = ws; ws += (size_t)B_ * C_ * NPBLK; // 262144
    float* psq   = ws; ws += (size_t)B_ * C_ * NPBLK;
    float* w1s   = ws; ws += (size_t)B_ * C_ * C_;    // 1M
    float* w2s   = ws; ws += (size_t)B_ * C_ * C_;
    float* h     = ws;                                // 67,108,864 (256 MB)

    const int gemm_blocks = B_ * (C_ / 16) * NPBLK;   // 16*16*64 = 16384

    // layer 1
    stats_x<<<B_ * C_, 256, 0, stream>>>(x, mean1, rstd1);
    scale_w<<<B_ * C_, 256, 0, stream>>>(w1, b1, mean1, rstd1, w1s, beff1);
    gemm16<<<gemm_blocks, 256, 0, stream>>>(x, w1s, beff1, h, psum, psq, 1);
    // layer 2
    stats_part<<<B_ * C_, 64, 0, stream>>>(psum, psq, mean2, rstd2);
    scale_w<<<B_ * C_, 256, 0, stream>>>(w2, b2, mean2, rstd2, w2s, beff2);
    gemm16<<<gemm_blocks, 256, 0, stream>>>(h, w2s, beff2, out, nullptr, nullptr, 0);
}